// QuanvLayer_40931038331593
// MI455X (gfx1250) — compile-verified
//
#include <hip/hip_runtime.h>
#include <math.h>

// Problem constants (fixed by the reference):
//   x: [64, 2, 512, 512] fp32, weights: [1,4] fp32, out: [64, 4, 256, 256] fp32
namespace {
constexpr int Bn = 64;
constexpr int H  = 512;
constexpr int W  = 512;
constexpr int NH = 256;   // H / 2
constexpr int NW = 256;   // W / 2
constexpr int THREADS = 256;                    // 8 wave32s per block
constexpr int TOTAL   = Bn * NH * (NW / 2);     // one thread handles 2 patches
constexpr int PLANE   = NH * NW;                // one output plane
}

// Native clang vectors: accepted by __builtin_nontemporal_{load,store},
// lower to global_load_b128 / global_store_b64 with TH=NT.
typedef __attribute__((ext_vector_type(4))) float vf4;
typedef __attribute__((ext_vector_type(2))) float vf2;

// One 2x2 patch: a* = real channel elems (r,c),(r,c+1),(r+1,c),(r+1,c+1); i* = imag.
// theta_k = (m_k - min)/(max - min + 1e-8) * pi + w_k ; c_k = cos(theta_k)
// out = [c1*c2*c3, c0*c1, c0*c1*c2, c0*c1*c2*c3]
__device__ __forceinline__ void quanv_patch(
    float a0, float a1, float a2, float a3,
    float i0, float i1, float i2, float i3,
    float w0, float w1, float w2, float w3,
    float& o0, float& o1, float& o2, float& o3)
{
    float m0 = __builtin_amdgcn_sqrtf(fmaf(a0, a0, i0 * i0));
    float m1 = __builtin_amdgcn_sqrtf(fmaf(a1, a1, i1 * i1));
    float m2 = __builtin_amdgcn_sqrtf(fmaf(a2, a2, i2 * i2));
    float m3 = __builtin_amdgcn_sqrtf(fmaf(a3, a3, i3 * i3));
    float lo = fminf(fminf(m0, m1), fminf(m2, m3));
    float hi = fmaxf(fmaxf(m0, m1), fmaxf(m2, m3));
    // s = pi / (range + 1e-8); v_rcp_f32 is plenty accurate for this scale factor
    float s  = 3.14159265358979323846f * __builtin_amdgcn_rcpf((hi - lo) + 1e-8f);
    float c0 = __cosf(fmaf(m0 - lo, s, w0));   // -> v_cos_f32 (native trans pipe)
    float c1 = __cosf(fmaf(m1 - lo, s, w1));
    float c2 = __cosf(fmaf(m2 - lo, s, w2));
    float c3 = __cosf(fmaf(m3 - lo, s, w3));
    o1 = c0 * c1;
    o2 = o1 * c2;
    o3 = o2 * c3;
    o0 = c1 * c2 * c3;
}

__global__ __launch_bounds__(THREADS)
void quanv_kernel(const float* __restrict__ x,
                  const float* __restrict__ wts,
                  float* __restrict__ out)
{
    const int idx = blockIdx.x * THREADS + threadIdx.x;   // grid sized exactly
    const int pw2 = idx & (NW / 2 - 1);                   // pair-of-patches along width
    const int ph  = (idx >> 7) & (NH - 1);                // patch row
    const int b   = idx >> 15;                            // batch

    // Uniform address -> scalarized to s_load_b128 by the compiler.
    const float w0 = wts[0], w1 = wts[1], w2 = wts[2], w3 = wts[3];

    const float* __restrict__ xr = x + (size_t)(b * 2 + 0) * (H * W); // real plane
    const float* __restrict__ xi = x + (size_t)(b * 2 + 1) * (H * W); // imag plane
    const int row = ph * 2;
    const int col = pw2 * 4;                               // 4 input cols = 2 patches

    // WGP-scope prefetch (locality 3): warm the next row-pair into all cache levels.
    __builtin_prefetch(xr + (size_t)(row + 2) * W + col, 0, 3);
    __builtin_prefetch(xi + (size_t)(row + 2) * W + col, 0, 3);

    // Four fully-coalesced 128-bit loads per thread; TH=NT — input is a one-shot
    // stream, don't let it occupy cache residency after use.
    const vf4 r0 = __builtin_nontemporal_load((const vf4*)(xr + (size_t)row       * W + col));
    const vf4 r1 = __builtin_nontemporal_load((const vf4*)(xr + (size_t)(row + 1) * W + col));
    const vf4 s0 = __builtin_nontemporal_load((const vf4*)(xi + (size_t)row       * W + col));
    const vf4 s1 = __builtin_nontemporal_load((const vf4*)(xi + (size_t)(row + 1) * W + col));

    float oA0, oA1, oA2, oA3, oB0, oB1, oB2, oB3;
    // Patch A: cols (col, col+1); ravel order (r,c),(r,c+1),(r+1,c),(r+1,c+1)
    quanv_patch(r0.x, r0.y, r1.x, r1.y, s0.x, s0.y, s1.x, s1.y,
                w0, w1, w2, w3, oA0, oA1, oA2, oA3);
    // Patch B: cols (col+2, col+3)
    quanv_patch(r0.z, r0.w, r1.z, r1.w, s0.z, s0.w, s1.z, s1.w,
                w0, w1, w2, w3, oB0, oB1, oB2, oB3);

    // out[b, q, ph, pw2*2 .. +1] as 64-bit stores per plane (8B aligned, coalesced).
    // TH=NT stores — output is written once and never re-read by this dispatch.
    float* o = out + ((size_t)(b * 4) * NH + ph) * NW + pw2 * 2;
    vf2 q0 = {oA0, oB0};
    vf2 q1 = {oA1, oB1};
    vf2 q2 = {oA2, oB2};
    vf2 q3 = {oA3, oB3};
    __builtin_nontemporal_store(q0, (vf2*)(o));
    __builtin_nontemporal_store(q1, (vf2*)(o + PLANE));
    __builtin_nontemporal_store(q2, (vf2*)(o + 2 * PLANE));
    __builtin_nontemporal_store(q3, (vf2*)(o + 3 * PLANE));
}

extern "C" void kernel_launch(void* const* d_in, const int* in_sizes, int n_in,
                              void* d_out, int out_size, void* d_ws, size_t ws_size,
                              hipStream_t stream) {
    const float* x   = (const float*)d_in[0];   // [64,2,512,512] fp32
    const float* wts = (const float*)d_in[1];   // [1,4] fp32
    float* out       = (float*)d_out;           // [64,4,256,256] fp32

    const int blocks = TOTAL / THREADS;         // 2,097,152 / 256 = 8192, exact
    quanv_kernel<<<blocks, THREADS, 0, stream>>>(x, wts, out);
}